// Bounding_Box_Detector_54829552501322
// MI455X (gfx1250) — compile-verified
//
#include <hip/hip_runtime.h>
#include <math.h>

#define Bn    2
#define NCLS  3            // foreground classes
#define BC    (Bn*NCLS)    // 6 independent problems
#define NPT   8192         // H*W points per problem
#define R2    9.0f

typedef __attribute__((ext_vector_type(2))) float v2f;
typedef __attribute__((ext_vector_type(8))) float v8f;

// ---------------- Stage 1: classification mask, |p|^2, zero scratch ----------------
__global__ void prep_kernel(const float* __restrict__ seg, const float* __restrict__ lidar,
                            float* __restrict__ rsq, unsigned* __restrict__ mask,
                            unsigned* __restrict__ keep, float* __restrict__ sums,
                            float* __restrict__ cnt) {
  int t = blockIdx.x * blockDim.x + threadIdx.x;
  if (t >= BC * NPT) return;
  int bc = t / NPT, n = t % NPT;
  int b = bc / NCLS, cls = bc % NCLS + 1;
  const float* sb = seg + (size_t)b * 4 * NPT;
  float v0 = sb[n], v1 = sb[NPT + n], v2 = sb[2 * NPT + n], v3 = sb[3 * NPT + n];
  int arg = 0; float mx = v0;                 // jnp.argmax: first occurrence of max
  if (v1 > mx) { mx = v1; arg = 1; }
  if (v2 > mx) { mx = v2; arg = 2; }
  if (v3 > mx) { mx = v3; arg = 3; }
  mask[t] = (arg == cls) ? 1u : 0u;
  keep[t] = 0u;
  cnt[t] = 0.f;
  sums[2 * t] = 0.f; sums[2 * t + 1] = 0.f;
  if (bc % NCLS == 0) {                        // once per batch
    float x = lidar[(size_t)b * 2 * NPT + n];
    float y = lidar[(size_t)b * 2 * NPT + NPT + n];
    rsq[b * NPT + n] = x * x + y * y;
  }
}

// ---------------- Stage 2: greedy radius NMS (sequential semantics) ----------------
// One block per (b,c). Tile pre-filter in parallel against compacted kept list in
// LDS, then wave 0 resolves in-tile index order with lane-parallel checks (wave32).
__global__ void __launch_bounds__(1024) nms_kernel(const float* __restrict__ lidar,
                                                   const unsigned* __restrict__ mask,
                                                   unsigned* __restrict__ keep) {
  __shared__ float2  s_kept[NPT];     // compacted kept centers (64 KB, CDNA5 LDS is 320 KB)
  __shared__ float   s_tx[1024];
  __shared__ float   s_ty[1024];
  __shared__ unsigned s_cand[1024];
  __shared__ int     s_cnt;
  int bc = blockIdx.x;
  int b  = bc / NCLS;
  int tid = threadIdx.x;
  int lane = tid & 31;
  const float* px = lidar + (size_t)b * 2 * NPT;
  const float* py = px + NPT;
  if (tid == 0) s_cnt = 0;
  __syncthreads();
  for (int tile = 0; tile < NPT / 1024; ++tile) {
    int i = tile * 1024 + tid;
    float x = px[i], y = py[i];
    unsigned m = mask[bc * NPT + i];
    int cend = s_cnt;                  // kept so far (all have smaller index)
    bool sup = false;
    for (int j = 0; j < cend; ++j) {
      float2 k = s_kept[j];
      float dx = x - k.x, dy = y - k.y;
      if (dx * dx + dy * dy < R2) { sup = true; break; }
    }
    s_tx[tid] = x; s_ty[tid] = y;
    s_cand[tid] = (m && !sup) ? 1u : 0u;
    __syncthreads();
    if (tid < 32) {                    // wave 0: in-order resolution within tile
      int cnt  = s_cnt;                // uniform across lanes
      int base = cnt;
      for (int t = 0; t < 1024; ++t) {
        if (s_cand[t]) {               // uniform branch (same LDS word)
          float xt = s_tx[t], yt = s_ty[t];
          bool s2 = false;
          for (int j = base + lane; j < cnt; j += 32) {
            float dx = xt - s_kept[j].x, dy = yt - s_kept[j].y;
            if (dx * dx + dy * dy < R2) s2 = true;
          }
          unsigned bal = __builtin_amdgcn_ballot_w32(s2);
          if (bal == 0u) {
            if (lane == 0) {
              s_kept[cnt] = make_float2(xt, yt);
              keep[bc * NPT + tile * 1024 + t] = 1u;
            }
            cnt++;                     // stays wave-uniform
          }
        }
      }
      if (lane == 0) s_cnt = cnt;
    }
    __syncthreads();
  }
}

// ---------------- Stage 3: nearest kept center via V_WMMA_F32_16X16X4_F32 ----------------
// d2(i,j) = |pi|^2 + |pj|^2 - 2*(pi . pj); cross term is a K=2 GEMM on the f32
// 16x16x4 WMMA (K slots 2,3 zero). A: lanes 0-15 carry rows M=0..15 (K=0,1 in two
// VGPRs); lanes 16-31 carry K=2,3 = 0. B: lanes 0-15 carry rows K=0,1 across cols;
// lanes 16-31 (K=2,3) = 0. C/D: slot r, lanes 0-15 -> (M=r, N=lane), lanes 16-31 ->
// (M=r+8, N=lane-16). Masking is folded into the column |pj|^2 term (3e38 for
// non-kept centers; best init 1e37 so they never win and argmin of an empty set is 0),
// keeping the inner loop fully branchless (v_cndmask, no saveexec).
__global__ void __launch_bounds__(256) assign_kernel(const float* __restrict__ lidar,
                                                     const float* __restrict__ rsq,
                                                     const unsigned* __restrict__ keep,
                                                     int* __restrict__ assign) {
  const int ROWS_PER_BLOCK = 128;              // 8 waves * 16 rows
  int blocksPerBC = NPT / ROWS_PER_BLOCK;      // 64
  int bc = blockIdx.x / blocksPerBC;
  int rowBase = (blockIdx.x % blocksPerBC) * ROWS_PER_BLOCK;
  int b = bc / NCLS;
  int wave = threadIdx.x >> 5;
  int lane = threadIdx.x & 31;
  int half = lane >> 4;
  int l16  = lane & 15;
  int i0 = rowBase + wave * 16;
  const float* px = lidar + (size_t)b * 2 * NPT;
  const float* py = px + NPT;
  const float* rs = rsq + b * NPT;
  const unsigned* kp = keep + bc * NPT;

  // A fragment: every lane loads (valid address), K=2,3 half selected to zero.
  float axl = px[i0 + l16], ayl = py[i0 + l16];
  v2f a;
  a[0] = half ? 0.f : axl;
  a[1] = half ? 0.f : ayl;

  float rowR[8];
#pragma unroll
  for (int r = 0; r < 8; ++r) rowR[r] = rs[i0 + r + 8 * half];

  float best[8]; int bestj[8];
#pragma unroll
  for (int r = 0; r < 8; ++r) { best[r] = 1.0e37f; bestj[r] = 0; }

#pragma unroll 4
  for (int j0 = 0; j0 < NPT; j0 += 16) {
    int jc = j0 + l16;
    float    colR = rs[jc];
    unsigned colK = kp[jc];
    float    bx   = px[jc], by = py[jc];
    float colRB = colK ? colR : 3.0e38f;       // mask folded into the |pj|^2 term
    v2f bb;
    bb[0] = half ? 0.f : bx;
    bb[1] = half ? 0.f : by;
    v8f acc = {};
    acc = __builtin_amdgcn_wmma_f32_16x16x4_f32(false, a, false, bb,
                                                (short)0, acc, false, false);
#pragma unroll
    for (int r = 0; r < 8; ++r) {
      float d2 = fmaf(-2.0f, acc[r], rowR[r] + colRB);
      bool c = d2 < best[r];                   // strict <: first j wins ties
      best[r]  = c ? d2 : best[r];
      bestj[r] = c ? jc : bestj[r];
    }
  }
  // min/argmin reduce within each 16-lane half (xor offsets 1..8 stay in-half)
#pragma unroll
  for (int r = 0; r < 8; ++r) {
    float bd = best[r]; int bj = bestj[r];
    for (int off = 8; off > 0; off >>= 1) {
      float od = __shfl_xor(bd, off, 32);
      int   oj = __shfl_xor(bj, off, 32);
      bool take = (od < bd) || (od == bd && oj < bj);
      bd = take ? od : bd;
      bj = take ? oj : bj;
    }
    if (l16 == 0) assign[bc * NPT + i0 + r + 8 * half] = bj;
  }
}

// ---------------- Stage 4: segment sums (global f32 atomics) ----------------
__global__ void scatter_kernel(const float* __restrict__ lidar, const unsigned* __restrict__ mask,
                               const int* __restrict__ assign, float* __restrict__ sums,
                               float* __restrict__ cnt) {
  int t = blockIdx.x * blockDim.x + threadIdx.x;
  if (t >= BC * NPT) return;
  if (!mask[t]) return;
  int bc = t / NPT, n = t % NPT, b = bc / NCLS;
  float x = lidar[(size_t)b * 2 * NPT + n];
  float y = lidar[(size_t)b * 2 * NPT + NPT + n];
  int a = assign[t];
  atomicAdd(&sums[(bc * NPT + a) * 2],     x);
  atomicAdd(&sums[(bc * NPT + a) * 2 + 1], y);
  atomicAdd(&cnt[bc * NPT + a], 1.0f);
}

// ---------------- Stage 5: final centers*keep and keep output ----------------
__global__ void final_kernel(const float* __restrict__ lidar, const unsigned* __restrict__ keep,
                             const float* __restrict__ sums, const float* __restrict__ cnt,
                             float* __restrict__ out) {
  int t = blockIdx.x * blockDim.x + threadIdx.x;
  if (t >= BC * NPT) return;
  int bc = t / NPT, n = t % NPT, b = bc / NCLS;
  unsigned k = keep[t];
  float cx = 0.f, cy = 0.f;
  if (k) {
    float ct = cnt[t];
    if (ct > 0.f) {
      float inv = 1.f / fmaxf(ct, 1.0f);
      cx = sums[2 * t] * inv; cy = sums[2 * t + 1] * inv;
    } else {
      cx = lidar[(size_t)b * 2 * NPT + n];
      cy = lidar[(size_t)b * 2 * NPT + NPT + n];
    }
  }
  out[2 * t]     = cx;
  out[2 * t + 1] = cy;
  out[2 * BC * NPT + t] = k ? 1.0f : 0.0f;   // keep, concatenated after centers
}

extern "C" void kernel_launch(void* const* d_in, const int* in_sizes, int n_in,
                              void* d_out, int out_size, void* d_ws, size_t ws_size,
                              hipStream_t stream) {
  const float* seg   = (const float*)d_in[0];   // [B,4,H,W]
  const float* lidar = (const float*)d_in[1];   // [B,2,H,W]
  float* out = (float*)d_out;

  char* ws = (char*)d_ws;
  float*    rsq    = (float*)ws;    ws += (size_t)Bn * NPT * sizeof(float);
  unsigned* mask   = (unsigned*)ws; ws += (size_t)BC * NPT * sizeof(unsigned);
  unsigned* keep   = (unsigned*)ws; ws += (size_t)BC * NPT * sizeof(unsigned);
  int*      assign = (int*)ws;      ws += (size_t)BC * NPT * sizeof(int);
  float*    sums   = (float*)ws;    ws += (size_t)BC * NPT * 2 * sizeof(float);
  float*    cnt    = (float*)ws;

  int total = BC * NPT;
  prep_kernel   <<<(total + 255) / 256, 256, 0, stream>>>(seg, lidar, rsq, mask, keep, sums, cnt);
  nms_kernel    <<<BC, 1024, 0, stream>>>(lidar, mask, keep);
  assign_kernel <<<BC * (NPT / 128), 256, 0, stream>>>(lidar, rsq, keep, assign);
  scatter_kernel<<<(total + 255) / 256, 256, 0, stream>>>(lidar, mask, assign, sums, cnt);
  final_kernel  <<<(total + 255) / 256, 256, 0, stream>>>(lidar, keep, sums, cnt, out);
}